// EquivGNNEncoder_51762945851636
// MI455X (gfx1250) — compile-verified
//
#include <hip/hip_runtime.h>
#include <math.h>

// ---------------------------------------------------------------------------
// CDNA5 (gfx1250) wave32 FP32 WMMA helpers.
// V_WMMA_F32_16X16X4_F32: A = 16x4 f32 (2 VGPR), B = 4x16 f32 (2 VGPR),
// C/D = 16x16 f32 (8 VGPR).
// Layout (ISA 7.12.2): lane t, m/n = t&15, khalf = t>>4:
//   A: {A[m][2*khalf], A[m][2*khalf+1]}
//   B: {B[2*khalf][n], B[2*khalf+1][n]}
//   D: reg r holds D[r + 8*khalf][n]
// ---------------------------------------------------------------------------
typedef float v2f __attribute__((ext_vector_type(2)));
typedef float v8f __attribute__((ext_vector_type(8)));

__device__ __forceinline__ v8f wmma4(v2f a, v2f b, v8f c) {
  return __builtin_amdgcn_wmma_f32_16x16x4_f32(false, a, false, b, (short)0, c,
                                               false, false);
}

#define INV_SQRT3_F 0.5773502691896258f
#define SQRT3_F     1.7320508075688772f
#define C_SCALAR_F  0.14433756729740643f   // 1/sqrt(48)
#define C_VEC_F     (0.25f * INV_SQRT3_F)  // sqrt(3/48) * 1/sqrt(3)
#define INV_S32_F   0.17677669529663687f   // 1/sqrt(32)
#define INV_S16_F   0.25f                  // 1/sqrt(16)

// ---------------------------------------------------------------------------
// s[n][u] = (emb[z[n]] @ W_s2n)[u] / sqrt(32)
// ---------------------------------------------------------------------------
__global__ __launch_bounds__(256) void embed_kernel(
    const int* __restrict__ z, const float* __restrict__ emb,
    const float* __restrict__ Ws2n, float* __restrict__ s, int N) {
  int idx = blockIdx.x * blockDim.x + threadIdx.x;
  if (idx >= N * 32) return;
  int n = idx >> 5, u = idx & 31;
  const float* erow = emb + (size_t)z[n] * 32;
  float acc = 0.f;
#pragma unroll
  for (int k = 0; k < 32; ++k) acc += erow[k] * Ws2n[k * 32 + u];
  s[idx] = acc * INV_S32_F;
}

// ---------------------------------------------------------------------------
// Edge message kernel (persistent waves):
//  - weights staged once per block into LDS, transposed so each WMMA B
//    operand is one aligned ds_load_b64
//  - one wave32 per 16-edge tile, all 4 TP paths as FP32 WMMA chains
//  - edge list is sorted by dst, so scatter uses run-length-merged atomics
// ---------------------------------------------------------------------------
__global__ __launch_bounds__(256) void edge_msg_kernel(
    const float* __restrict__ s, const float* __restrict__ v,
    const float* __restrict__ pos, const int* __restrict__ ei,
    const float* __restrict__ gW1, const float* __restrict__ gW2,
    const float* __restrict__ gW3, const float* __restrict__ gW4,
    float* __restrict__ a_s, float* __restrict__ a_v, int E, int ntiles) {
  // Transposed weights: lW*t[n][k] = W*[k][n]  (B operand = 2 consecutive k)
  __shared__ float lW1t[32 * 32];
  __shared__ float lW2t[16 * 32];
  __shared__ float lW3t[16 * 16];
  __shared__ float lW4t[32 * 16];

  for (int i = threadIdx.x; i < 1024; i += 256) {
    int k = i >> 5, n = i & 31;
    lW1t[n * 32 + k] = gW1[i];
  }
  for (int i = threadIdx.x; i < 512; i += 256) {
    int k = i >> 4, n = i & 15;
    lW2t[n * 32 + k] = gW2[i];
  }
  for (int i = threadIdx.x; i < 256; i += 256) {
    int k = i >> 4, n = i & 15;
    lW3t[n * 16 + k] = gW3[i];
  }
  for (int i = threadIdx.x; i < 512; i += 256) {
    int k = i >> 5, n = i & 31;
    lW4t[n * 16 + k] = gW4[i];
  }
  __syncthreads();

  const int wave = threadIdx.x >> 5;
  const int lane = threadIdx.x & 31;
  const int m = lane & 15;   // A row (edge) / B,D column
  const int kh = lane >> 4;  // K-half select
  const int wstride = gridDim.x * 8;

  for (int tile = blockIdx.x * 8 + wave; tile < ntiles; tile += wstride) {
    const int ebase = tile * 16;

    // Lanes 0..15 own one edge each: geometry + indices.
    int sn = -1, dn = 0;
    float shx = 0.f, shy = 0.f, shz = 0.f;
    {
      int eg = ebase + lane;
      if (lane < 16 && eg < E) {
        sn = ei[eg];
        dn = ei[E + eg];
        float dx = pos[(size_t)dn * 3 + 0] - pos[(size_t)sn * 3 + 0];
        float dy = pos[(size_t)dn * 3 + 1] - pos[(size_t)sn * 3 + 1];
        float dz = pos[(size_t)dn * 3 + 2] - pos[(size_t)sn * 3 + 2];
        float inv = SQRT3_F * rsqrtf(dx * dx + dy * dy + dz * dz);
        shx = dx * inv; shy = dy * inv; shz = dz * inv;
      }
    }
    // Wave-synchronous broadcast of edge-m data to both K-half lane groups.
    int src_m = __shfl(sn, m);
    float mshx = __shfl(shx, m);
    float mshy = __shfl(shy, m);
    float mshz = __shfl(shz, m);
    float vmask = (src_m >= 0) ? 1.0f : 0.0f;
    int sidx = (src_m >= 0) ? src_m : 0;
    const float* srow = s + (size_t)sidx * 32;
    const float* vrow = v + (size_t)sidx * 48;

    // A operand: ss tile (16 edges x 32), 8 K-chunks of 4.
    v2f a_ss[8];
#pragma unroll
    for (int c = 0; c < 8; ++c) {
      int k = 4 * c + 2 * kh;
      v2f a = {srow[k] * vmask, srow[k + 1] * vmask};
      a_ss[c] = a;
    }
    // vs elements this lane feeds into A operands: u in {4c+2kh, 4c+2kh+1}.
    float vu[8][3];
#pragma unroll
    for (int j = 0; j < 8; ++j) {
      int u = 4 * (j >> 1) + 2 * kh + (j & 1);
      vu[j][0] = vrow[u * 3 + 0] * vmask;
      vu[j][1] = vrow[u * 3 + 1] * vmask;
      vu[j][2] = vrow[u * 3 + 2] * vmask;
    }
    // d[e][u] = (1/sqrt3) * vs[e][u] . sh1[e]   (1o x 1o -> 0e)
    float dvals[8];
#pragma unroll
    for (int j = 0; j < 8; ++j)
      dvals[j] =
          INV_SQRT3_F * (vu[j][0] * mshx + vu[j][1] * mshy + vu[j][2] * mshz);

    v8f accS0 = {}, accS1 = {}, accT = {};
    v8f accV0 = {}, accV1 = {}, accV2 = {};

    // d @ W4 (16x16 x 16x32) -> scalar-message chain
#pragma unroll
    for (int c = 0; c < 4; ++c) {
      int k = 4 * c + 2 * kh;
      v2f a = {dvals[2 * c], dvals[2 * c + 1]};
      v2f b0 = {lW4t[m * 16 + k], lW4t[m * 16 + k + 1]};
      v2f b1 = {lW4t[(16 + m) * 16 + k], lW4t[(16 + m) * 16 + k + 1]};
      accS0 = wmma4(a, b0, accS0);
      accS1 = wmma4(a, b1, accS1);
    }
    // ss @ W1 (16x32 x 32x32) accumulated into same chain
#pragma unroll
    for (int c = 0; c < 8; ++c) {
      int k = 4 * c + 2 * kh;
      v2f b0 = {lW1t[m * 32 + k], lW1t[m * 32 + k + 1]};
      v2f b1 = {lW1t[(16 + m) * 32 + k], lW1t[(16 + m) * 32 + k + 1]};
      accS0 = wmma4(a_ss[c], b0, accS0);
      accS1 = wmma4(a_ss[c], b1, accS1);
    }
    // t = ss @ W2 (16x32 x 32x16), scaled per-row by sh1 at the scatter
#pragma unroll
    for (int c = 0; c < 8; ++c) {
      int k = 4 * c + 2 * kh;
      v2f b = {lW2t[m * 32 + k], lW2t[m * 32 + k + 1]};
      accT = wmma4(a_ss[c], b, accT);
    }
    // vs_i @ W3 (16x16 x 16x16) per Cartesian component
#pragma unroll
    for (int c = 0; c < 4; ++c) {
      int k = 4 * c + 2 * kh;
      v2f b = {lW3t[m * 16 + k], lW3t[m * 16 + k + 1]};
      v2f a0 = {vu[2 * c][0], vu[2 * c + 1][0]};
      v2f a1 = {vu[2 * c][1], vu[2 * c + 1][1]};
      v2f a2 = {vu[2 * c][2], vu[2 * c + 1][2]};
      accV0 = wmma4(a0, b, accV0);
      accV1 = wmma4(a1, b, accV1);
      accV2 = wmma4(a2, b, accV2);
    }

    // Per-row edge metadata for this lane's 8 D rows (er = r + 8*kh).
    int dns[8];
    float exs[8], eys[8], ezs[8];
#pragma unroll
    for (int r = 0; r < 8; ++r) {
      int er = r + 8 * kh;
      dns[r] = __shfl(dn, er);
      exs[r] = __shfl(shx, er);
      eys[r] = __shfl(shy, er);
      ezs[r] = __shfl(shz, er);
    }

    // Scatter with run-length merge: edge list is sorted by dst, so
    // consecutive rows usually share a destination node.
    float sS0 = 0.f, sS1 = 0.f, sV0 = 0.f, sV1 = 0.f, sV2 = 0.f;
    int prev = dns[0];
#pragma unroll
    for (int r = 0; r < 8; ++r) {
      if (dns[r] != prev) {
        atomicAdd(&a_s[(size_t)prev * 32 + m], C_SCALAR_F * sS0);
        atomicAdd(&a_s[(size_t)prev * 32 + 16 + m], C_SCALAR_F * sS1);
        atomicAdd(&a_v[(size_t)prev * 48 + m * 3 + 0], C_VEC_F * sV0);
        atomicAdd(&a_v[(size_t)prev * 48 + m * 3 + 1], C_VEC_F * sV1);
        atomicAdd(&a_v[(size_t)prev * 48 + m * 3 + 2], C_VEC_F * sV2);
        sS0 = sS1 = sV0 = sV1 = sV2 = 0.f;
        prev = dns[r];
      }
      sS0 += accS0[r];
      sS1 += accS1[r];
      float t = accT[r];
      sV0 += t * exs[r] + accV0[r];
      sV1 += t * eys[r] + accV1[r];
      sV2 += t * ezs[r] + accV2[r];
    }
    atomicAdd(&a_s[(size_t)prev * 32 + m], C_SCALAR_F * sS0);
    atomicAdd(&a_s[(size_t)prev * 32 + 16 + m], C_SCALAR_F * sS1);
    atomicAdd(&a_v[(size_t)prev * 48 + m * 3 + 0], C_VEC_F * sV0);
    atomicAdd(&a_v[(size_t)prev * 48 + m * 3 + 1], C_VEC_F * sV1);
    atomicAdd(&a_v[(size_t)prev * 48 + m * 3 + 2], C_VEC_F * sV2);
  }
}

// ---------------------------------------------------------------------------
// Node-wise e3nn Linear + ReLU + residual (tiny: scalar VALU is fine)
// ---------------------------------------------------------------------------
__global__ __launch_bounds__(256) void node_update_kernel(
    const float* __restrict__ a_s, const float* __restrict__ a_v,
    const float* __restrict__ Ws, const float* __restrict__ Wv,
    float* __restrict__ s, float* __restrict__ v, int N) {
  int idx = blockIdx.x * blockDim.x + threadIdx.x;
  if (idx >= N * 80) return;
  int n = idx / 80, f = idx % 80;
  if (f < 32) {
    const float* row = a_s + (size_t)n * 32;
    float acc = 0.f;
#pragma unroll
    for (int u = 0; u < 32; ++u) acc += row[u] * Ws[u * 32 + f];
    acc *= INV_S32_F;
    if (acc > 0.f) s[(size_t)n * 32 + f] += acc;
  } else {
    int g = f - 32, w = g / 3, i = g % 3;
    const float* row = a_v + (size_t)n * 48;
    float acc = 0.f;
#pragma unroll
    for (int u = 0; u < 16; ++u) acc += row[u * 3 + i] * Wv[u * 16 + w];
    acc *= INV_S16_F;
    if (acc > 0.f) v[(size_t)n * 48 + w * 3 + i] += acc;
  }
}

// ---------------------------------------------------------------------------
// Sum-pool per graph (batch is contiguous blocks of natoms nodes)
// ---------------------------------------------------------------------------
__global__ __launch_bounds__(256) void pool_kernel(
    const float* __restrict__ s, const float* __restrict__ v,
    float* __restrict__ hg, int B, int natoms) {
  int idx = blockIdx.x * blockDim.x + threadIdx.x;
  if (idx >= B * 80) return;
  int g = idx / 80, c = idx % 80;
  int n0 = g * natoms;
  float acc = 0.f;
  if (c < 32) {
    for (int a = 0; a < natoms; ++a) acc += s[(size_t)(n0 + a) * 32 + c];
  } else {
    int cc = c - 32;
    for (int a = 0; a < natoms; ++a) acc += v[(size_t)(n0 + a) * 48 + cc];
  }
  hg[idx] = acc;
}

// ---------------------------------------------------------------------------
// Dense FP32-WMMA GEMM:  C = act(A(MxK) @ B(KxN) + bias), one 16x16 C tile
// per wave32. K must be a multiple of 4 (80 and 256 here).
// ---------------------------------------------------------------------------
__global__ __launch_bounds__(256) void gemm_bias_act_kernel(
    const float* __restrict__ A, const float* __restrict__ B,
    const float* __restrict__ bias, float* __restrict__ C, int M, int N,
    int K, int relu) {
  int wave = threadIdx.x >> 5;
  int lane = threadIdx.x & 31;
  int tiles_n = N >> 4;
  int gt = blockIdx.x * 8 + wave;
  int tm = gt / tiles_n, tn = gt % tiles_n;
  if (tm * 16 >= M) return;  // wave-uniform

  int m = lane & 15, kh = lane >> 4;
  const float* arow = A + (size_t)(tm * 16 + m) * K;
  const float* bcol = B + tn * 16 + m;
  v8f acc = {};
#pragma unroll 4
  for (int k = 0; k < K; k += 4) {
    int kk = k + 2 * kh;
    v2f a = {arow[kk], arow[kk + 1]};
    v2f b = {bcol[(size_t)kk * N], bcol[(size_t)(kk + 1) * N]};
    acc = wmma4(a, b, acc);
  }
  float bv = bias[tn * 16 + m];
#pragma unroll
  for (int r = 0; r < 8; ++r) {
    int row = tm * 16 + r + 8 * kh;
    float val = acc[r] + bv;
    if (relu) val = fmaxf(val, 0.f);
    C[(size_t)row * N + tn * 16 + m] = val;
  }
}

// ---------------------------------------------------------------------------
// Host orchestration
// ---------------------------------------------------------------------------
extern "C" void kernel_launch(void* const* d_in, const int* in_sizes, int n_in,
                              void* d_out, int out_size, void* d_ws,
                              size_t ws_size, hipStream_t stream) {
  const float* pos = (const float*)d_in[0];
  const int* z = (const int*)d_in[1];
  const int* ei = (const int*)d_in[3];
  const float* emb = (const float*)d_in[5];
  const float* Ws2n = (const float*)d_in[6];
  const float* W1 = (const float*)d_in[7];
  const float* W2 = (const float*)d_in[8];
  const float* W3 = (const float*)d_in[9];
  const float* W4 = (const float*)d_in[10];
  const float* Ws = (const float*)d_in[11];
  const float* Wv = (const float*)d_in[12];
  const float* Wr1 = (const float*)d_in[13];
  const float* br1 = (const float*)d_in[14];
  const float* Wr2 = (const float*)d_in[15];
  const float* br2 = (const float*)d_in[16];

  const int N = in_sizes[0] / 3;
  const int E = in_sizes[3] / 2;
  const int B = out_size / 128;          // latent = 128
  const int natoms = N / B;
  const int L = in_sizes[7] / (32 * 32); // layers

  float* ws = (float*)d_ws;
  float* s_buf = ws;
  float* v_buf = s_buf + (size_t)N * 32;
  float* as_buf = v_buf + (size_t)N * 48;
  float* av_buf = as_buf + (size_t)N * 32;
  float* hg_buf = av_buf + (size_t)N * 48;
  float* h1_buf = hg_buf + (size_t)B * 80;

  hipMemsetAsync(v_buf, 0, (size_t)N * 48 * sizeof(float), stream);
  embed_kernel<<<(N * 32 + 255) / 256, 256, 0, stream>>>(z, emb, Ws2n, s_buf,
                                                         N);

  const int ntiles = (E + 15) / 16;
  int eblocks = (ntiles + 7) / 8;
  if (eblocks > 2048) eblocks = 2048;  // persistent waves, grid-stride
  for (int l = 0; l < L; ++l) {
    hipMemsetAsync(as_buf, 0, (size_t)N * 32 * sizeof(float), stream);
    hipMemsetAsync(av_buf, 0, (size_t)N * 48 * sizeof(float), stream);
    edge_msg_kernel<<<eblocks, 256, 0, stream>>>(
        s_buf, v_buf, pos, ei, W1 + (size_t)l * 32 * 32,
        W2 + (size_t)l * 32 * 16, W3 + (size_t)l * 16 * 16,
        W4 + (size_t)l * 16 * 32, as_buf, av_buf, E, ntiles);
    node_update_kernel<<<(N * 80 + 255) / 256, 256, 0, stream>>>(
        as_buf, av_buf, Ws + (size_t)l * 32 * 32, Wv + (size_t)l * 16 * 16,
        s_buf, v_buf, N);
  }

  pool_kernel<<<(B * 80 + 255) / 256, 256, 0, stream>>>(s_buf, v_buf, hg_buf,
                                                        B, natoms);

  const int t1 = (B / 16) * (256 / 16);
  gemm_bias_act_kernel<<<(t1 + 7) / 8, 256, 0, stream>>>(hg_buf, Wr1, br1,
                                                         h1_buf, B, 256, 80, 1);
  const int t2 = (B / 16) * (128 / 16);
  gemm_bias_act_kernel<<<(t2 + 7) / 8, 256, 0, stream>>>(
      h1_buf, Wr2, br2, (float*)d_out, B, 128, 256, 0);
}